// GaussianVectorQuantizer_22600117911979
// MI455X (gfx1250) — compile-verified
//
#include <hip/hip_runtime.h>
#include <math.h>

// GaussianVectorQuantizer (train path) fused for gfx1250 / MI455X.
// GEMM1 (z @ cb^T) and GEMM2 (enc @ cb) use V_WMMA_F32_16X16X4_F32.
// Gumbel tile is prefetched (global_prefetch_b8) under the GEMM1 WMMA chain.

typedef float v2f __attribute__((ext_vector_type(2)));
typedef float v8f __attribute__((ext_vector_type(8)));

#define K_CODES 512
#define DIMZ    64
#define WH      1024            // W*H = 32*32
#define BS      32
#define NROWS   (BS * WH)       // 32768
#define LPAD    516             // padded LDS row stride (floats): bank-conflict-free
#define TEMP_INV 2.0f           // 1/TEMPERATURE

__device__ __forceinline__ float waveReduceSum(float v) {
#pragma unroll
  for (int m = 1; m < 32; m <<= 1) v += __shfl_xor(v, m, 32);
  return v;
}
__device__ __forceinline__ float waveReduceMax(float v) {
#pragma unroll
  for (int m = 1; m < 32; m <<= 1) v = fmaxf(v, __shfl_xor(v, m, 32));
  return v;
}

// ws layout (floats): [0]=weight, [1]=kld_discrete acc, [2]=kld_cont acc,
// [3..515)=avg_probs acc (512), [515..1027)=codebook row norms (512)

__global__ __launch_bounds__(512)
void vq_init_kernel(const float* __restrict__ var_q,
                    const float* __restrict__ codebook,
                    float* __restrict__ ws) {
  const int tid = threadIdx.x;
  if (tid == 0) {
    ws[0] = 0.5f / fmaxf(var_q[0], 1e-10f);
    ws[1] = 0.0f;
    ws[2] = 0.0f;
  }
  float s = 0.f;
  const float* row = codebook + tid * DIMZ;
#pragma unroll
  for (int c = 0; c < DIMZ; ++c) s += row[c] * row[c];
  ws[3 + tid]   = 0.0f;   // avg_probs accumulator
  ws[515 + tid] = s;      // ||c_k||^2
}

__global__ __launch_bounds__(32)
void vq_main_kernel(const float* __restrict__ z,
                    const float* __restrict__ codebook,
                    const float* __restrict__ gumbel,
                    float* __restrict__ out,
                    float* __restrict__ ws) {
  __shared__ float lbuf[16][LPAD];   // logits, later unnormalized encodings
  __shared__ float rowScale[16];     // 1/sum(exp) for gumbel softmax per row

  const int lane = threadIdx.x;      // one wave32 per block
  const int nl   = lane & 15;        // N-column within 16x16 tile / A row m
  const int half = lane >> 4;
  const int n0   = blockIdx.x * 16;  // first flattened row of this tile

  const float weight = ws[0];
  const float w2     = 2.0f * weight;
  const float* cnorm = ws + 515;

  // Prefetch this block's gumbel slice (16 rows x 512 f32 = 32KB) so the
  // HBM fetch overlaps the GEMM1 WMMA chain. 32 lanes x 128B lines = 4KB
  // per prefetch instruction; 8 instructions cover the tile.
  {
    const char* gbase = reinterpret_cast<const char*>(gumbel + (size_t)n0 * K_CODES)
                        + (size_t)lane * 128;
#pragma unroll
    for (int i = 0; i < 8; ++i)
      __builtin_prefetch(gbase + (size_t)i * 4096, 0, 3);
  }

  // ---------- Stage 1: logits' = weight*(2*z.c - |c|^2) via f32 WMMA ----------
  // z_flat[n][c] = z[b][c][wh], b=n>>10, wh=n&1023  (channels-last flatten)
  const int b   = n0 >> 10;
  const int wh0 = n0 & (WH - 1);
  const float* zrow = z + (size_t)b * DIMZ * WH + (wh0 + nl); // row m = nl
  // Preload all 16 A fragments (16x64 tile of z), reused across 32 col tiles.
  v2f afrag[16];
#pragma unroll
  for (int ks = 0; ks < 16; ++ks) {
    const int c0 = 4 * ks + 2 * half;               // k index within A
    afrag[ks].x = zrow[(size_t)c0 * WH];
    afrag[ks].y = zrow[(size_t)(c0 + 1) * WH];
  }
  const float2* cb2 = reinterpret_cast<const float2*>(codebook);
  for (int t = 0; t < 32; ++t) {
    const int col0 = t * 16;
    v8f acc = {};
#pragma unroll
    for (int ks = 0; ks < 16; ++ks) {
      // B[k][n] = codebook[col0+n][4*ks + 2*half + {0,1}] (contiguous pair)
      float2 bp = cb2[(size_t)(col0 + nl) * (DIMZ / 2) + (2 * ks + half)];
      v2f bfrag; bfrag.x = bp.x; bfrag.y = bp.y;
      acc = __builtin_amdgcn_wmma_f32_16x16x4_f32(
          false, afrag[ks], false, bfrag, (short)0, acc, false, false);
    }
    const float bias = weight * cnorm[col0 + nl];
#pragma unroll
    for (int j = 0; j < 8; ++j)                 // D: lane holds rows j+8*half
      lbuf[j + 8 * half][col0 + nl] = w2 * acc[j] - bias;
  }
  __syncthreads();

  // ---------- Stage 2+3: softmax stats + gumbel-softmax encodings ----------
  float kld_acc = 0.f;
  float kc_acc  = 0.f;
  float avgAcc[16];
#pragma unroll
  for (int i = 0; i < 16; ++i) avgAcc[i] = 0.f;

  for (int r = 0; r < 16; ++r) {
    const int n = n0 + r;
    float lv[16];
#pragma unroll
    for (int i = 0; i < 16; ++i) lv[i] = lbuf[r][lane + 32 * i];
    float M1 = -INFINITY;
#pragma unroll
    for (int i = 0; i < 16; ++i) M1 = fmaxf(M1, lv[i]);
    M1 = waveReduceMax(M1);

    float ev[16]; float E = 0.f, Bl = 0.f;
#pragma unroll
    for (int i = 0; i < 16; ++i) {
      const float e = __expf(lv[i] - M1);
      ev[i] = e; E += e; Bl += e * lv[i];
    }
    E  = waveReduceSum(E);
    Bl = waveReduceSum(Bl);
    const float invE = 1.0f / E;
    // sum_k p*log p = (sum e*l)/E - M1 - log(E)
    kld_acc += Bl * invE - M1 - __logf(E);
#pragma unroll
    for (int i = 0; i < 16; ++i) avgAcc[i] += ev[i] * invE;

    // gumbel noise (coalesced: 32 lanes read consecutive cols of row n)
    const float* gu = gumbel + (size_t)n * K_CODES;
    float sv[16]; float M2 = -INFINITY;
#pragma unroll
    for (int i = 0; i < 16; ++i) {
      const float u = gu[lane + 32 * i];
      const float g = -__logf(-__logf(u + 1e-10f) + 1e-10f);
      const float s = TEMP_INV * (lv[i] + g);
      sv[i] = s; M2 = fmaxf(M2, s);
    }
    M2 = waveReduceMax(M2);
    float E2 = 0.f;
#pragma unroll
    for (int i = 0; i < 16; ++i) {
      const float e2 = __expf(sv[i] - M2);
      E2 += e2;
      lbuf[r][lane + 32 * i] = e2;     // overwrite logits with unnorm enc
    }
    E2 = waveReduceSum(E2);
    if (lane == 0) rowScale[r] = 1.0f / E2;
  }
  __syncthreads();

  // ---------- Stage 4: z_q = (enc_unnorm @ codebook) * rowScale via WMMA ------
  v8f acc2[4] = {};
  for (int ks = 0; ks < 128; ++ks) {
    const int kA = 4 * ks + 2 * half;
    const float2 ap = *reinterpret_cast<const float2*>(&lbuf[nl][kA]); // A row m=nl
    v2f a2; a2.x = ap.x; a2.y = ap.y;
#pragma unroll
    for (int ct = 0; ct < 4; ++ct) {
      const int c0 = ct * 16;
      v2f b2;
      b2.x = codebook[(size_t)kA * DIMZ + c0 + nl];        // B[k][c]
      b2.y = codebook[(size_t)(kA + 1) * DIMZ + c0 + nl];
      acc2[ct] = __builtin_amdgcn_wmma_f32_16x16x4_f32(
          false, a2, false, b2, (short)0, acc2[ct], false, false);
    }
  }
  // Epilogue: normalize, write z_to_decoder (bs,dim_z,W,H), accumulate MSE.
#pragma unroll
  for (int ct = 0; ct < 4; ++ct) {
    const int c = ct * 16 + nl;
    const size_t base = ((size_t)(b * DIMZ + c)) * WH + wh0;
#pragma unroll
    for (int j = 0; j < 8; ++j) {
      const int m = j + 8 * half;
      const float zq = acc2[ct][j] * rowScale[m];
      const size_t idx = base + m;
      const float d = z[idx] - zq;
      kc_acc += d * d;
      out[idx] = zq;
    }
  }

  // ---------- Stage 5: flush accumulators ----------
  kc_acc = waveReduceSum(kc_acc);
  float* g_avg = ws + 3;
#pragma unroll
  for (int i = 0; i < 16; ++i)
    unsafeAtomicAdd(&g_avg[lane + 32 * i], avgAcc[i]);
  if (lane == 0) {
    unsafeAtomicAdd(&ws[1], kld_acc);
    unsafeAtomicAdd(&ws[2], kc_acc);
  }
}

__global__ __launch_bounds__(512)
void vq_final_kernel(float* __restrict__ out, const float* __restrict__ ws) {
  __shared__ float red[512];
  const int tid = threadIdx.x;
  const float a = ws[3 + tid] * (1.0f / (float)NROWS);
  red[tid] = a * __logf(a + 1e-7f);
  __syncthreads();
  for (int s = 256; s > 0; s >>= 1) {
    if (tid < s) red[tid] += red[tid + s];
    __syncthreads();
  }
  if (tid == 0) {
    const float weight = ws[0];
    const float loss = ws[1] / (float)BS + ws[2] * weight / (float)BS;
    out[(size_t)NROWS * DIMZ]     = loss;
    out[(size_t)NROWS * DIMZ + 1] = __expf(-red[0]);
  }
}

extern "C" void kernel_launch(void* const* d_in, const int* in_sizes, int n_in,
                              void* d_out, int out_size, void* d_ws, size_t ws_size,
                              hipStream_t stream) {
  const float* z        = (const float*)d_in[0];
  const float* var_q    = (const float*)d_in[1];
  const float* codebook = (const float*)d_in[2];
  const float* gumbel   = (const float*)d_in[3];
  // d_in[4], d_in[5]: flg_train=1, flg_quant_det=0 -> train path implemented.
  float* out = (float*)d_out;
  float* ws  = (float*)d_ws;

  vq_init_kernel<<<1, 512, 0, stream>>>(var_q, codebook, ws);
  vq_main_kernel<<<NROWS / 16, 32, 0, stream>>>(z, codebook, gumbel, out, ws);
  vq_final_kernel<<<1, 512, 0, stream>>>(out, ws);
}